// HyperbolicMAELoss_48387101557437
// MI455X (gfx1250) — compile-verified
//
#include <hip/hip_runtime.h>
#include <hip/hip_bf16.h>
#include <math.h>

// Problem constants (fixed by the reference setup)
#define NROWS 4096
#define DDIM  128
#define NT    (NROWS / 16)                    // 256 tile rows/cols
#define TOTAL_TILES (NT * (NT + 1) / 2)       // 32896 lower-tri tiles (incl diag)
#define WAVES_PER_BLOCK 8
#define NBLOCKS ((TOTAL_TILES + WAVES_PER_BLOCK - 1) / WAVES_PER_BLOCK)  // 4112
#define CLAMP_TOL 1e-06

typedef __attribute__((ext_vector_type(2))) float v2f;
typedef __attribute__((ext_vector_type(8))) float v8f;

// ws layout (doubles): ws[0] = loss accumulator, ws[1 .. NROWS] = s[i]
__global__ void hyp_prep_kernel(const float* __restrict__ outm,
                                double* __restrict__ ws) {
    int i = blockIdx.x * blockDim.x + threadIdx.x;
    if (i == 0) ws[0] = 0.0;                  // zero accumulator (runs before main)
    if (i < NROWS) {
        const float* row = outm + (size_t)i * DDIM;
        double acc = 0.0;
#pragma unroll 8
        for (int k = 0; k < DDIM; ++k) {
            double v = (double)row[k];
            acc += v * v;
        }
        ws[1 + i] = sqrt(1.0 + acc);
    }
}

// One wave32 per 16x16 tile of B = s⊗s - out@out^T, lower triangle only.
__global__ void __launch_bounds__(256) hyp_gram_loss_kernel(
        const float* __restrict__ outm,
        const float* __restrict__ tgt,
        double* __restrict__ ws) {
    const int lane = threadIdx.x & 31;
    const int wave = threadIdx.x >> 5;
    const int w = blockIdx.x * WAVES_PER_BLOCK + wave;   // wave-uniform
    if (w >= TOTAL_TILES) return;                        // uniform -> EXEC stays all-1s

    // Decode linear lower-triangular tile index -> (bi, bj), bi >= bj
    int bi = (int)((sqrt(8.0 * (double)w + 1.0) - 1.0) * 0.5);
    while ((bi + 1) * (bi + 2) / 2 <= w) ++bi;
    while (bi * (bi + 1) / 2 > w) --bi;
    const int bj = w - bi * (bi + 1) / 2;
    const int row0 = bi * 16;
    const int col0 = bj * 16;

    // V_WMMA_F32_16X16X4_F32 operand layout:
    //   A (16x4): lane L -> M = L&15; half = L>>4 selects K pair {2*half, 2*half+1}
    //   B (4x16): lane L -> N = L&15; same half/K-pair striping
    // Gram: A[m][k] = out[row0+m][k], B[k][n] = out[col0+n][k]
    // -> both fragments are the same aligned 8-byte gather on different row blocks.
    const int mn   = lane & 15;
    const int half = lane >> 4;
    const float* pa = outm + (size_t)(row0 + mn) * DDIM + 2 * half;
    const float* pb = outm + (size_t)(col0 + mn) * DDIM + 2 * half;

    v8f c = {};
#pragma unroll
    for (int k0 = 0; k0 < DDIM; k0 += 4) {
        v2f a = *(const v2f*)(pa + k0);   // 8B aligned: row*128 + 2*half + k0 even
        v2f b = *(const v2f*)(pb + k0);
        // (neg_a, A, neg_b, B, c_mod, C, reuse_a, reuse_b)
        c = __builtin_amdgcn_wmma_f32_16x16x4_f32(
                false, a, false, b, (short)0, c, false, false);
    }

    // C/D layout: VGPR r holds M = r + 8*half, N = lane&15
    const double* s = ws + 1;
    double acc = 0.0;
#pragma unroll
    for (int r = 0; r < 8; ++r) {
        const int i = row0 + r + 8 * half;
        const int j = col0 + mn;
        if (i > j) {                       // strict lower triangle
            double B = s[i] * s[j] - (double)c[r];
            double dist;
            if (fabs(B - 1.0) < CLAMP_TOL) {
                dist = 0.0;
            } else {
                dist = log(B + sqrt(B * B - 1.0));   // acosh(B), B >= 1 here
            }
            double t = (double)tgt[(size_t)i * NROWS + j];
            acc += fabs(dist - t);
        }
    }

    // wave32 butterfly reduction, then one f64 atomic per wave
#pragma unroll
    for (int off = 16; off > 0; off >>= 1)
        acc += __shfl_down(acc, off, 32);
    if (lane == 0)
        atomicAdd(ws, acc);
}

__global__ void hyp_finalize_kernel(const double* __restrict__ ws,
                                    float* __restrict__ out) {
    if (threadIdx.x == 0 && blockIdx.x == 0) {
        out[0] = (float)(ws[0] / ((double)NROWS * (double)(NROWS - 1)));
    }
}

extern "C" void kernel_launch(void* const* d_in, const int* in_sizes, int n_in,
                              void* d_out, int out_size, void* d_ws, size_t ws_size,
                              hipStream_t stream) {
    const float* outm = (const float*)d_in[0];   // [4096,128] fp32
    const float* tgt  = (const float*)d_in[1];   // [4096,4096] fp32
    float* dout = (float*)d_out;                 // scalar
    double* ws  = (double*)d_ws;                 // [1 + 4096] doubles

    hyp_prep_kernel<<<(NROWS + 255) / 256, 256, 0, stream>>>(outm, ws);
    hyp_gram_loss_kernel<<<NBLOCKS, 256, 0, stream>>>(outm, tgt, ws);
    hyp_finalize_kernel<<<1, 64, 0, stream>>>(ws, dout);
}